// LocalGrouper_41274635714630
// MI455X (gfx1250) — compile-verified
//
#include <hip/hip_runtime.h>
#include <cstdint>
#include <cstddef>

// ---------------------------------------------------------------------------
// LocalGrouper for MI455X (gfx1250, wave32, WGP, 320KB LDS, WMMA)
//
// Pipeline (all on `stream`, deterministic):
//   0) zero_f32        : clear per-batch sum-of-squares accumulator
//   1) fps_kernel      : farthest point sampling, 1 block/batch, cloud in LDS
//                        -> fps_idx[B,S] (ws), new_xyz (d_out[0:81920])
//   2) knn_kernel      : distance tiles via V_WMMA_F32_16X16X4_F32 into
//                        16x4096 f32 LDS tile (256KB), exact top-32 selection
//                        -> knn_idx[B,S,K] (ws)
//   3) stats_kernel    : per-(b,s,c) mean over K + batch sum of sq-deviations
//   4) writeout_kernel : normalize + affine + anchor concat -> d_out tail
// ---------------------------------------------------------------------------

#define BATCH    16
#define NPTS     4096
#define CXYZ     5
#define CH       128
#define CC       133          // CH + CXYZ
#define S_GROUPS 1024
#define KNN_K    32
#define COUT     261          // CC + CH
#define EPSV     1e-5f

typedef float v2f __attribute__((ext_vector_type(2)));
typedef float v8f __attribute__((ext_vector_type(8)));

// ---------------------------------------------------------------------------
__global__ void zero_f32(float* __restrict__ p, int n) {
  int i = blockIdx.x * blockDim.x + threadIdx.x;
  if (i < n) p[i] = 0.0f;
}

// ---------------------------------------------------------------------------
// Farthest point sampling. One 1024-thread block per batch. Entire cloud in
// LDS (SoA, 5*4096 f32 = 80KB); each thread owns 4 points' running min-dist
// in registers. Per step: broadcast centroid from LDS, update, argmax via
// wave32 shuffles + cross-wave LDS reduce. Tie-break: smaller index (jnp.argmax).
__global__ __launch_bounds__(1024) void fps_kernel(const float* __restrict__ xyz,
                                                   int* __restrict__ fps_idx,
                                                   float* __restrict__ out_newxyz) {
  extern __shared__ float sm[];                 // [CXYZ][NPTS] + reduce scratch
  const int b = blockIdx.x;
  const int t = threadIdx.x;
  const int lane = t & 31, wave = t >> 5;
  const float* xb = xyz + (size_t)b * NPTS * CXYZ;
  float* redD = sm + CXYZ * NPTS;
  int*   redI = (int*)(redD + 32);
  int*   resI = redI + 32;

  #pragma unroll
  for (int j = 0; j < 4; ++j) {
    int n = t + j * 1024;
    #pragma unroll
    for (int c = 0; c < CXYZ; ++c) sm[c * NPTS + n] = xb[n * CXYZ + c];
  }
  float dist[4] = {1e10f, 1e10f, 1e10f, 1e10f};
  __syncthreads();

  int cur = 0;
  if (t == 0) {
    fps_idx[b * S_GROUPS] = 0;
    #pragma unroll
    for (int c = 0; c < CXYZ; ++c)
      out_newxyz[((size_t)b * S_GROUPS) * CXYZ + c] = sm[c * NPTS];
  }

  for (int i = 1; i < S_GROUPS; ++i) {
    float cx[CXYZ];
    #pragma unroll
    for (int c = 0; c < CXYZ; ++c) cx[c] = sm[c * NPTS + cur];   // broadcast

    float best = -1.0f; int bidx = 0;
    #pragma unroll
    for (int j = 0; j < 4; ++j) {
      int n = t + j * 1024;
      float d = 0.0f;
      #pragma unroll
      for (int c = 0; c < CXYZ; ++c) {
        float df = sm[c * NPTS + n] - cx[c];
        d += df * df;
      }
      dist[j] = fminf(dist[j], d);
      if (j == 0 || dist[j] > best) { best = dist[j]; bidx = n; }  // n ascends: first-max kept
    }
    #pragma unroll
    for (int off = 16; off; off >>= 1) {
      float od = __shfl_xor(best, off, 32);
      int   oi = __shfl_xor(bidx, off, 32);
      if (od > best || (od == best && oi < bidx)) { best = od; bidx = oi; }
    }
    if (lane == 0) { redD[wave] = best; redI[wave] = bidx; }
    __syncthreads();
    if (wave == 0) {
      float bd = redD[lane]; int bi = redI[lane];
      #pragma unroll
      for (int off = 16; off; off >>= 1) {
        float od = __shfl_xor(bd, off, 32);
        int   oi = __shfl_xor(bi, off, 32);
        if (od > bd || (od == bd && oi < bi)) { bd = od; bi = oi; }
      }
      if (lane == 0) *resI = bi;
    }
    __syncthreads();
    cur = *resI;
    if (t == 0) {
      fps_idx[b * S_GROUPS + i] = cur;
      #pragma unroll
      for (int c = 0; c < CXYZ; ++c)
        out_newxyz[((size_t)b * S_GROUPS + i) * CXYZ + c] = sm[c * NPTS + cur];
    }
  }
}

// ---------------------------------------------------------------------------
// KNN. One block per (batch, 16-query tile): 1024 blocks, 256 threads (8 waves).
// Distances via f32 WMMA with norms folded into K-padding:
//   A row(m) = [-2q0..-2q4, ||q||^2, 1, 0]   (16x8, two 16x4 k-chunks)
//   B col(n) = [ p0.. p4,   1, ||p||^2, 0]   (8x16)
//   D[m][n]  = ||q||^2 + ||p||^2 - 2 q.p
// Full 16x4096 f32 distance tile lives in 256KB LDS (CDNA5 WGP LDS).
// A-frag layout (ISA 7.12.2): lanes 0-15 hold (K0,K1), lanes 16-31 (K2,K3),
// M = lane%16; B symmetric with N = lane%16; D VGPR r: M=r (+8 for hi half).
// The tile loop has a compile-time trip count so the backend emits a uniform
// scalar loop: EXEC stays all-1s at every WMMA (ISA 7.12 requirement).
__global__ __launch_bounds__(256) void knn_kernel(const float* __restrict__ xyz,
                                                  const int* __restrict__ fps_idx,
                                                  int* __restrict__ knn_idx) {
  extern __shared__ float dlds[];               // [16][NPTS] f32 = 256KB
  const int blk  = blockIdx.x;
  const int b    = blk >> 6;                    // 64 query tiles per batch
  const int qbase = (blk & 63) << 4;
  const int t = threadIdx.x, lane = t & 31, wave = t >> 5;
  const int half = lane >> 4, l16 = lane & 15;
  const float* xb = xyz + (size_t)b * NPTS * CXYZ;

  // A fragment (same queries for every tile in this block)
  const int q  = qbase + l16;
  const int qi = fps_idx[b * S_GROUPS + q];
  float qc[CXYZ];
  #pragma unroll
  for (int c = 0; c < CXYZ; ++c) qc[c] = xb[qi * CXYZ + c];
  const float qn = qc[0]*qc[0] + qc[1]*qc[1] + qc[2]*qc[2] + qc[3]*qc[3] + qc[4]*qc[4];
  v2f a0, a1;
  a0.x = half ? -2.0f * qc[2] : -2.0f * qc[0];
  a0.y = half ? -2.0f * qc[3] : -2.0f * qc[1];
  a1.x = half ? 1.0f          : -2.0f * qc[4];
  a1.y = half ? 0.0f          : qn;

  constexpr int NTILES = NPTS / 16;             // 256
  constexpr int NITER  = NTILES / 8;            // 32 tiles per wave, uniform
  #pragma unroll 1
  for (int it = 0; it < NITER; ++it) {
    const int tile = it * 8 + wave;
    const int p = tile * 16 + l16;
    // prefetch next tile's coords (global_prefetch_b8) to hide VMEM latency
    if (it + 1 < NITER)
      __builtin_prefetch(xb + (size_t)(p + 128) * CXYZ, 0, 0);

    float pc[CXYZ];
    #pragma unroll
    for (int c = 0; c < CXYZ; ++c) pc[c] = xb[p * CXYZ + c];
    const float pn = pc[0]*pc[0] + pc[1]*pc[1] + pc[2]*pc[2] + pc[3]*pc[3] + pc[4]*pc[4];
    v2f b0, b1;
    b0.x = half ? pc[2] : pc[0];
    b0.y = half ? pc[3] : pc[1];
    b1.x = half ? pn    : pc[4];
    b1.y = half ? 0.0f  : 1.0f;

    v8f acc = {};
    acc = __builtin_amdgcn_wmma_f32_16x16x4_f32(false, a0, false, b0,
                                                (short)0, acc, false, false);
    acc = __builtin_amdgcn_wmma_f32_16x16x4_f32(false, a1, false, b1,
                                                (short)0, acc, false, false);
    #pragma unroll
    for (int r = 0; r < 8; ++r) {
      const int m = r + (half << 3);
      dlds[m * NPTS + tile * 16 + l16] = acc[r];
    }
  }
  __syncthreads();

  // Exact top-32 per query: each wave owns 2 queries; 32 rounds of wave-wide
  // min(+index tie-break) over the LDS row, invalidating the winner.
  // Each lane scans 4 consecutive floats per chunk -> ds_load_b128.
  for (int sel = wave * 2; sel < wave * 2 + 2; ++sel) {
    float* row = dlds + sel * NPTS;
    for (int round = 0; round < KNN_K; ++round) {
      float best = 3.0e38f; int bidx = NPTS;
      #pragma unroll 1
      for (int chunk = 0; chunk < NPTS / 128; ++chunk) {
        const int nbase = chunk * 128 + lane * 4;
        const float4 d4 = *(const float4*)(row + nbase);
        if (d4.x < best) { best = d4.x; bidx = nbase;     }  // ascending n:
        if (d4.y < best) { best = d4.y; bidx = nbase + 1; }  // strict < keeps
        if (d4.z < best) { best = d4.z; bidx = nbase + 2; }  // first (lowest)
        if (d4.w < best) { best = d4.w; bidx = nbase + 3; }  // index on ties
      }
      #pragma unroll
      for (int off = 16; off; off >>= 1) {
        float od = __shfl_xor(best, off, 32);
        int   oi = __shfl_xor(bidx, off, 32);
        if (od < best || (od == best && oi < bidx)) { best = od; bidx = oi; }
      }
      if (lane == 0) {
        knn_idx[((size_t)(b * S_GROUPS) + qbase + sel) * KNN_K + round] = bidx;
        row[bidx] = 3.0e38f;                    // same-wave DS ops are in-order
      }
    }
  }
}

// ---------------------------------------------------------------------------
// Per-(b,s,c) mean over K + per-batch sum of squared deviations.
// Sum(g-mu)^2 = Sum g^2 - K*mu^2. Channel-major threads -> coalesced gathers.
__global__ __launch_bounds__(256) void stats_kernel(const float* __restrict__ xyz,
                                                    const float* __restrict__ points,
                                                    const int* __restrict__ knn_idx,
                                                    float* __restrict__ means,
                                                    float* __restrict__ ssq) {
  const int blk = blockIdx.x;                   // (b,s)
  const int b = blk >> 10;
  const int t = threadIdx.x;
  __shared__ int   sIdx[KNN_K];
  __shared__ float redS[8];
  if (t < KNN_K) sIdx[t] = knn_idx[(size_t)blk * KNN_K + t];
  __syncthreads();

  float dev2 = 0.0f;
  if (t < CC) {
    float sum = 0.0f, sum2 = 0.0f;
    for (int k = 0; k < KNN_K; ++k) {
      const int id = sIdx[k];
      const float v = (t < CH) ? points[((size_t)b * NPTS + id) * CH + t]
                               : xyz[((size_t)b * NPTS + id) * CXYZ + (t - CH)];
      sum += v; sum2 += v * v;
    }
    const float mean = sum * (1.0f / KNN_K);
    means[(size_t)blk * CC + t] = mean;
    dev2 = sum2 - (float)KNN_K * mean * mean;
  }
  #pragma unroll
  for (int off = 16; off; off >>= 1) dev2 += __shfl_xor(dev2, off, 32);
  if ((t & 31) == 0) redS[t >> 5] = dev2;
  __syncthreads();
  if (t == 0) {
    float tot = 0.0f;
    #pragma unroll
    for (int w = 0; w < 8; ++w) tot += redS[w];
    atomicAdd(&ssq[b], tot);
  }
}

// ---------------------------------------------------------------------------
// Normalize + affine + anchor concat. One block per (b,s); bandwidth-bound.
__global__ __launch_bounds__(256) void writeout_kernel(const float* __restrict__ xyz,
                                                       const float* __restrict__ points,
                                                       const float* __restrict__ alpha,
                                                       const float* __restrict__ beta,
                                                       const int* __restrict__ fps_idx,
                                                       const int* __restrict__ knn_idx,
                                                       const float* __restrict__ means,
                                                       const float* __restrict__ ssq,
                                                       float* __restrict__ out) {
  const int blk = blockIdx.x;                   // (b,s)
  const int b = blk >> 10;
  const int t = threadIdx.x;
  __shared__ float sMean[CC], sAlpha[CC], sBeta[CC], sAnchor[CH];
  __shared__ int   sIdx[KNN_K];
  if (t < CC) {
    sMean[t]  = means[(size_t)blk * CC + t];
    sAlpha[t] = alpha[t];
    sBeta[t]  = beta[t];
  }
  if (t < KNN_K) sIdx[t] = knn_idx[(size_t)blk * KNN_K + t];
  const int fi = fps_idx[blk];
  if (t < CH) sAnchor[t] = points[((size_t)b * NPTS + fi) * CH + t];
  const float stdv = sqrtf(ssq[b] / (float)(S_GROUPS * KNN_K * CC - 1));  // ddof=1
  const float inv  = 1.0f / (stdv + EPSV);
  __syncthreads();

  float* ob = out + (size_t)BATCH * S_GROUPS * CXYZ + (size_t)blk * KNN_K * COUT;
  for (int k = 0; k < KNN_K; ++k) {
    const int id = sIdx[k];
    for (int c = t; c < COUT; c += 256) {
      float v;
      if (c < CC) {
        const float g = (c < CH) ? points[((size_t)b * NPTS + id) * CH + c]
                                 : xyz[((size_t)b * NPTS + id) * CXYZ + (c - CH)];
        v = sAlpha[c] * ((g - sMean[c]) * inv) + sBeta[c];
      } else {
        v = sAnchor[c - CC];
      }
      ob[(size_t)k * COUT + c] = v;
    }
  }
}

// ---------------------------------------------------------------------------
extern "C" void kernel_launch(void* const* d_in, const int* in_sizes, int n_in,
                              void* d_out, int out_size, void* d_ws, size_t ws_size,
                              hipStream_t stream) {
  (void)in_sizes; (void)n_in; (void)out_size; (void)ws_size;
  const float* xyz    = (const float*)d_in[0];   // [B,N,5]
  const float* points = (const float*)d_in[1];   // [B,N,128]
  const float* alpha  = (const float*)d_in[2];   // [133]
  const float* beta   = (const float*)d_in[3];   // [133]
  float* out = (float*)d_out;                    // [B,S,5] ++ [B,S,K,261]

  char* ws = (char*)d_ws;
  const size_t FPS_BYTES  = (size_t)BATCH * S_GROUPS * sizeof(int);            // 64KB
  const size_t KNN_BYTES  = (size_t)BATCH * S_GROUPS * KNN_K * sizeof(int);    // 2MB
  const size_t MEAN_BYTES = (size_t)BATCH * S_GROUPS * CC * sizeof(float);     // 8.7MB
  int*   fps   = (int*)ws;
  int*   knn   = (int*)(ws + FPS_BYTES);
  float* means = (float*)(ws + FPS_BYTES + KNN_BYTES);
  float* ssq   = (float*)(ws + FPS_BYTES + KNN_BYTES + MEAN_BYTES);

  zero_f32<<<1, 32, 0, stream>>>(ssq, BATCH);

  const size_t fpsLds = (size_t)CXYZ * NPTS * sizeof(float) + 32 * sizeof(float)
                      + 32 * sizeof(int) + sizeof(int);
  fps_kernel<<<BATCH, 1024, fpsLds, stream>>>(xyz, fps, out);

  const size_t knnLds = (size_t)16 * NPTS * sizeof(float);                     // 256KB
  knn_kernel<<<BATCH * (S_GROUPS / 16), 256, knnLds, stream>>>(xyz, fps, knn);

  stats_kernel<<<BATCH * S_GROUPS, 256, 0, stream>>>(xyz, points, knn, means, ssq);

  writeout_kernel<<<BATCH * S_GROUPS, 256, 0, stream>>>(xyz, points, alpha, beta,
                                                        fps, knn, means, ssq, out);
}